// NoisyKGate_20289425506607
// MI455X (gfx1250) — compile-verified
//
#include <hip/hip_runtime.h>
#include <stdint.h>

typedef __attribute__((ext_vector_type(2))) float v2f;
typedef __attribute__((ext_vector_type(4))) float v4f;
typedef __attribute__((ext_vector_type(8))) float v8f;
typedef __attribute__((ext_vector_type(4))) unsigned int u32x4;
typedef __attribute__((ext_vector_type(4))) int i32x4;
typedef __attribute__((ext_vector_type(8))) int i32x8;

#define D_DIM   2048
#define E_DIM   64
#define K_TOP   8
#define KC      64          // K-chunk per LDS stage
#define ROWS    128         // tokens per block (8 waves * 16)
#define XSTR    68          // padded LDS row stride (floats) -> conflict-free b64 A reads
#define M_TOTAL 16384       // B*S tokens

#if defined(__has_builtin)
#  if __has_builtin(__builtin_amdgcn_tensor_load_to_lds) && \
      __has_builtin(__builtin_amdgcn_s_wait_tensorcnt)
#    define USE_TDM 1
#  endif
#endif
#ifndef USE_TDM
#  define USE_TDM 0
#endif

#if USE_TDM
// Issue one TDM descriptor: load a 128-row x 64-col f32 tile (rows strided by
// D_DIM floats in global) into LDS, hardware-padding each 64-dword row by
// 4 dwords -> LDS row stride of 68 floats (bank-conflict-free for WMMA A reads).
__device__ __forceinline__ void tdm_load_x_tile(const float* gsrc, uint32_t lds_off)
{
    uint64_t ga = (uint64_t)(uintptr_t)gsrc;
    u32x4 g0;
    g0[0] = 1u;                                   // count=1 (valid user descriptor)
    g0[1] = lds_off;                              // lds_addr (bytes)
    g0[2] = (uint32_t)ga;                         // global_addr[31:0]
    g0[3] = (uint32_t)((ga >> 32) & 0x1FFFFFFu)   // global_addr[56:32]
          | (2u << 30);                           // type = 2 ("image")
    i32x8 g1;
    g1[0] = (int)((2u << 16)                      // data_size = 4 bytes
                | (1u << 20)                      // pad_enable
                | (5u << 22)                      // pad_interval: 2^(5+1)=64 dwords
                | (3u << 25));                    // pad_amount: 3+1 = 4 dwords
    g1[1] = (int)((uint32_t)D_DIM << 16);         // tensor_dim0[15:0] in [31:16]
    g1[2] = (int)((uint32_t)M_TOTAL << 16);       // tensor_dim1[15:0] in [31:16]
    g1[3] = (int)((uint32_t)KC << 16);            // tile_dim0 = 64 in [31:16]
    g1[4] = (int)ROWS;                            // tile_dim1 = 128 in [15:0]
    g1[5] = (int)D_DIM;                           // tensor_dim0_stride[31:0]
    g1[6] = 0;
    g1[7] = 0;
    i32x4 gz = {0, 0, 0, 0};
#if __clang_major__ >= 23
    i32x8 gz8 = {0, 0, 0, 0, 0, 0, 0, 0};
    __builtin_amdgcn_tensor_load_to_lds(g0, g1, gz, gz, gz8, 0);
#else
    __builtin_amdgcn_tensor_load_to_lds(g0, g1, gz, gz, 0);
#endif
}
#endif

__global__ __launch_bounds__(256) void noisyk_gate_kernel(
    const float* __restrict__ x,      // [M_TOTAL, D]
    const float* __restrict__ W,      // [D, E]
    const float* __restrict__ bias,   // [E]
    float* __restrict__ out_g,        // [M_TOTAL, 8]
    float* __restrict__ out_idx,      // [M_TOTAL, 8] (indices as float)
    float* __restrict__ out_scores)   // [M_TOTAL, E]
{
    __shared__ float xs[ROWS * XSTR];          // 34816 B : x tile (reused for top-k)
    __shared__ float wf[16 * 4 * 32 * 2];      // 16384 B : W chunk in B-fragment order

    const int tid  = threadIdx.x;
    const int wave = tid >> 5;
    const int lane = tid & 31;
    const int half = lane >> 4;        // 0: lanes 0-15, 1: lanes 16-31
    const int l16  = lane & 15;
    const int row0_blk = blockIdx.x * ROWS;
    const int row0     = row0_blk + wave * 16;

#if USE_TDM
    const uint32_t xs_lds = (uint32_t)(uintptr_t)(void*)xs;
#endif

    v8f acc0 = {}, acc1 = {}, acc2 = {}, acc3 = {};

    for (int k0 = 0; k0 < D_DIM; k0 += KC) {
#if USE_TDM
        // ---- stage x tile via Tensor Data Mover (async DMA, TENSORcnt) ----
        if (wave == 0)
            tdm_load_x_tile(x + (size_t)row0_blk * D_DIM + k0, xs_lds);
#else
        // ---- stage x tile: 128 rows x 64 cols, coalesced b128 loads ----
        #pragma unroll
        for (int p = 0; p < 8; ++p) {
            int flat = p * 256 + tid;
            int row  = flat >> 4;
            int c4   = (flat & 15) << 2;
            v4f v = *(const v4f*)(x + (size_t)(row0_blk + row) * D_DIM + k0 + c4);
            *(v4f*)(xs + row * XSTR + c4) = v;
        }
#endif
        // ---- stage W chunk pre-swizzled into WMMA B-fragment layout ----
        // fragment slot (s, t, lane): two floats = W[k0+4s+2*(lane/16)][t*16+lane%16]
        //                             and the next K row (per 16x16x4 f32 B layout)
        #pragma unroll
        for (int p = 0; p < 8; ++p) {
            int flat = p * 256 + tid;          // 2048 slots total
            int l = flat & 31;
            int t = (flat >> 5) & 3;
            int s = flat >> 7;                 // 0..15
            int col  = t * 16 + (l & 15);
            int krow = k0 + 4 * s + 2 * (l >> 4);
            v2f w;
            w.x = W[(size_t)krow * E_DIM + col];
            w.y = W[(size_t)(krow + 1) * E_DIM + col];
            *(v2f*)(wf + flat * 2) = w;
        }
#if USE_TDM
        if (wave == 0)
            __builtin_amdgcn_s_wait_tensorcnt(0);   // DMA complete before barrier
#endif
        __syncthreads();

        // ---- MACs: per K-step of 4 -> 1 A-frag load, 4 B-frag loads, 4 WMMAs ----
        const float* aptr = xs + (wave * 16 + l16) * XSTR + 2 * half;
        #pragma unroll
        for (int s = 0; s < 16; ++s) {
            v2f a  = *(const v2f*)(aptr + 4 * s);
            v2f b0 = *(const v2f*)(wf + ((s * 4 + 0) * 32 + lane) * 2);
            v2f b1 = *(const v2f*)(wf + ((s * 4 + 1) * 32 + lane) * 2);
            v2f b2 = *(const v2f*)(wf + ((s * 4 + 2) * 32 + lane) * 2);
            v2f b3 = *(const v2f*)(wf + ((s * 4 + 3) * 32 + lane) * 2);
            acc0 = __builtin_amdgcn_wmma_f32_16x16x4_f32(false, a, false, b0, (short)0, acc0, false, false);
            acc1 = __builtin_amdgcn_wmma_f32_16x16x4_f32(false, a, false, b1, (short)0, acc1, false, false);
            acc2 = __builtin_amdgcn_wmma_f32_16x16x4_f32(false, a, false, b2, (short)0, acc2, false, false);
            acc3 = __builtin_amdgcn_wmma_f32_16x16x4_f32(false, a, false, b3, (short)0, acc3, false, false);
        }
        __syncthreads();
    }

    // ---- epilogue: bias + sigmoid, emit scores, stash tile in LDS for top-k ----
    float bb[4];
    #pragma unroll
    for (int t = 0; t < 4; ++t) bb[t] = bias[t * 16 + l16];

    v8f accs[4] = {acc0, acc1, acc2, acc3};
    float* tile = xs + (wave * 16) * XSTR;     // reuse x staging area (wave-private rows)
    #pragma unroll
    for (int t = 0; t < 4; ++t) {
        int n = t * 16 + l16;
        #pragma unroll
        for (int j = 0; j < 8; ++j) {
            // C/D layout: VGPR j -> row M = j (lanes 0-15) or M = j+8 (lanes 16-31)
            float z = accs[t][j] + bb[t];
            float s = 1.0f / (1.0f + __expf(-z));
            int m = j + 8 * half;
            out_scores[(size_t)(row0 + m) * E_DIM + n] = s;
            tile[m * XSTR + n] = s;
        }
    }
    __syncthreads();

    // ---- top-8 per token: lanes 0-15 each own one token's 64 scores ----
    if (lane < 16) {
        const float* srow = xs + (wave * 16 + lane) * XSTR;  // conflict-free: bank = (4*lane+i)%64
        float pv = 3.402823466e38f;   // previous selected value
        int   pi = -1;                // previous selected index
        float topv[K_TOP]; int topi[K_TOP];
        float sum = 0.0f;
        for (int j = 0; j < K_TOP; ++j) {
            float bestv = -1.0f;      // sigmoid > 0 always
            int   besti = 0;
            for (int i = 0; i < E_DIM; ++i) {
                float v = srow[i];
                // eligible = not already selected (strictly below prev, or tied with larger index)
                bool elig   = (v < pv) || ((v == pv) && (i > pi));
                bool better = elig && (v > bestv);   // strict > keeps lowest-index tie-break
                bestv = better ? v : bestv;
                besti = better ? i : besti;
            }
            topv[j] = bestv; topi[j] = besti;
            sum += bestv;
            pv = bestv; pi = besti;
        }
        float inv = 1.0f / sum;
        size_t token = (size_t)(row0 + lane);
        #pragma unroll
        for (int j = 0; j < K_TOP; ++j) {
            out_g[token * K_TOP + j]   = topv[j] * inv;
            out_idx[token * K_TOP + j] = (float)topi[j];
        }
    }
}

extern "C" void kernel_launch(void* const* d_in, const int* in_sizes, int n_in,
                              void* d_out, int out_size, void* d_ws, size_t ws_size,
                              hipStream_t stream) {
    const float* x    = (const float*)d_in[0];   // [4,4096,2048] f32
    const float* W    = (const float*)d_in[1];   // [2048,64] f32
    const float* bias = (const float*)d_in[2];   // [64] f32
    // d_in[3] = k (==8), baked in at compile time

    float* out    = (float*)d_out;
    float* g      = out;                          // [16384,8]
    float* idx    = out + M_TOTAL * K_TOP;        // [16384,8]
    float* scores = out + 2 * M_TOTAL * K_TOP;    // [16384,64]

    dim3 grid(M_TOTAL / ROWS), block(256);
    hipLaunchKernelGGL(noisyk_gate_kernel, grid, block, 0, stream,
                       x, W, bias, g, idx, scores);
}